// VectorQuantizer_31894427140211
// MI455X (gfx1250) — compile-verified
//
#include <hip/hip_runtime.h>
#include <hip/hip_bf16.h>

// ---------------------------------------------------------------------------
// VectorQuantizer forward for MI455X (gfx1250, wave32).
//
// Shapes: inputs [16,3,256,256] f32 planar, embedding [256,3] f32, nc=16.
// Heavy stage (dnorm over all 256 embeddings x 1M pixels) is mapped onto
// V_WMMA_F32_16X16X4_F32: A = [x0,x1,x2,1] (16 pixels x K=4),
// B = [-2e0,-2e1,-2e2,||e||^2] (K=4 x 16 embeddings) so the WMMA directly
// produces ||e||^2 - 2 x.e per (pixel,k) tile element in f32 (reference-
// matching precision). Remaining per-element work: + ||x||^2, clamp, add.
// ---------------------------------------------------------------------------

typedef __attribute__((ext_vector_type(2))) float v2f;
typedef __attribute__((ext_vector_type(8))) float v8f;

#define BATCH 16
#define HW    65536      // 256*256
#define KEMB  256
#define NC    16
#define P1_CHUNKS 64     // pixel chunks per batch in phase 1

// ---------------------------------------------------------------- phase 0 ---
__global__ void vq_zero_kernel(float* __restrict__ p, int n) {
  int i = blockIdx.x * blockDim.x + threadIdx.x;
  if (i < n) p[i] = 0.0f;
}

// ---------------------------------------------------------------- phase 1 ---
// dnorm2[b][k] += sum over pixels of max(||x||^2 + ||e_k||^2 - 2 x.e_k, 0)
__global__ __launch_bounds__(256) void vq_dnorm2_kernel(
    const float* __restrict__ in, const float* __restrict__ emb,
    float* __restrict__ dnorm2) {
  const int b     = blockIdx.y;
  const int tid   = threadIdx.x;
  const int lane  = tid & 31;
  const int wave  = tid >> 5;
  const int lane15 = lane & 15;
  const bool lo   = lane < 16;

  // B-matrix tiles for all 256 embeddings: 16 tiles of 16, kept in VGPRs.
  // ISA 16x16x4 B layout: lanes 0-15 hold K-rows 0,1 (v0,v1) for N=lane;
  // lanes 16-31 hold K-rows 2,3.
  float b0r[16], b1r[16];
#pragma unroll
  for (int t = 0; t < 16; ++t) {
    int k = t * 16 + lane15;
    float e0 = emb[3 * k + 0];
    float e1 = emb[3 * k + 1];
    float e2c = emb[3 * k + 2];
    if (lo) { b0r[t] = -2.0f * e0;  b1r[t] = -2.0f * e1; }
    else    { b0r[t] = -2.0f * e2c; b1r[t] = e0 * e0 + e1 * e1 + e2c * e2c; }
  }

  float ksum[16];
#pragma unroll
  for (int t = 0; t < 16; ++t) ksum[t] = 0.0f;

  const float* base0 = in + (size_t)b * 3 * HW;
  const int pixPerBlock = HW / P1_CHUNKS;        // 1024
  const int pixPerWave  = pixPerBlock / 8;       // 128 -> 8 tiles of 16
  const int pix0 = blockIdx.x * pixPerBlock + wave * pixPerWave;

  for (int it = 0; it < pixPerWave / 16; ++it) {
    int p = pix0 + it * 16 + lane15;
    // A layout: lanes 0-15 -> (x0,x1), lanes 16-31 -> (x2, 1.0 bias)
    float a0 = lo ? base0[p] : base0[2 * HW + p];
    float a1 = lo ? base0[HW + p] : 1.0f;

    // ||x||^2 per pixel (exclude the bias 1.0 on hi lanes)
    float part = lo ? (a0 * a0 + a1 * a1) : (a0 * a0);
    float x2full = part + __shfl_xor(part, 16, 32);
    float x2row[8];
#pragma unroll
    for (int j = 0; j < 8; ++j) {
      int row = lo ? j : (8 + j);                // C rows this lane owns
      x2row[j] = __shfl(x2full, row, 32);        // ds_bpermute broadcast
    }

    v2f A; A.x = a0; A.y = a1;
#pragma unroll
    for (int t = 0; t < 16; ++t) {
      v2f Bv; Bv.x = b0r[t]; Bv.y = b1r[t];
      v8f c = {};
      // D = A x B : d[pixel,k] = ||e_k||^2 - 2 x.e_k   (f32, RNE)
      c = __builtin_amdgcn_wmma_f32_16x16x4_f32(
          false, A, false, Bv, (short)0, c, false, false);
#pragma unroll
      for (int j = 0; j < 8; ++j) {
        float d2 = x2row[j] + c[j];
        ksum[t] += fmaxf(d2, 0.0f);
      }
    }
  }

  // Combine the two half-wave row groups, then one atomic per (k) lane.
#pragma unroll
  for (int t = 0; t < 16; ++t) {
    float tot = ksum[t] + __shfl_xor(ksum[t], 16, 32);
    if (lo) atomicAdd(&dnorm2[b * KEMB + t * 16 + lane15], tot);
  }
}

// ---------------------------------------------------------------- phase 2 ---
// Per batch: dnorm = sqrt(dnorm2); iterative top-16 (largest value, ties ->
// lowest index, matching jax.lax.top_k); emit color_palette + ws palette.
__global__ __launch_bounds__(256) void vq_topk_kernel(
    const float* __restrict__ emb, const float* __restrict__ dnorm2,
    float* __restrict__ pal_ws /*[B][NC][4]*/,
    float* __restrict__ out_pal /*[B][NC][3]*/) {
  __shared__ float sval[256];
  __shared__ int   sidx[256];
  __shared__ int   winners[NC];
  const int b = blockIdx.x;
  const int tid = threadIdx.x;

  float myv = sqrtf(dnorm2[b * KEMB + tid]);
  bool taken = false;

  for (int iter = 0; iter < NC; ++iter) {
    sval[tid] = taken ? -1.0f : myv;   // dnorm >= 0 so -1 acts as -inf
    sidx[tid] = tid;
    __syncthreads();
    for (int s = 128; s > 0; s >>= 1) {
      if (tid < s) {
        float v2v = sval[tid + s]; int i2 = sidx[tid + s];
        float v1v = sval[tid];     int i1 = sidx[tid];
        if (v2v > v1v || (v2v == v1v && i2 < i1)) {
          sval[tid] = v2v; sidx[tid] = i2;
        }
      }
      __syncthreads();
    }
    if (tid == 0) winners[iter] = sidx[0];
    __syncthreads();
    if (tid == winners[iter]) taken = true;
    __syncthreads();
  }

  if (tid < NC) {
    int k = winners[tid];
    float e0 = emb[3 * k + 0], e1 = emb[3 * k + 1], e2c = emb[3 * k + 2];
    out_pal[(b * NC + tid) * 3 + 0] = e0;
    out_pal[(b * NC + tid) * 3 + 1] = e1;
    out_pal[(b * NC + tid) * 3 + 2] = e2c;
    pal_ws[(b * NC + tid) * 4 + 0] = e0;
    pal_ws[(b * NC + tid) * 4 + 1] = e1;
    pal_ws[(b * NC + tid) * 4 + 2] = e2c;
    pal_ws[(b * NC + tid) * 4 + 3] = 0.0f;
  }
}

// ---------------------------------------------------------------- phase 3 ---
// Nearest palette entry per pixel; write quant (3 planes) + one-hot (16
// planes) + per-block squared-error partial sums (deterministic reduction).
__global__ __launch_bounds__(256) void vq_assign_kernel(
    const float* __restrict__ in, const float* __restrict__ pal_ws,
    float* __restrict__ out_q, float* __restrict__ out_enc,
    float* __restrict__ blockSums) {
  __shared__ float pal[NC * 4];
  __shared__ float red[256];
  const int b = blockIdx.y;
  const int tid = threadIdx.x;
  if (tid < NC * 4) pal[tid] = pal_ws[b * NC * 4 + tid];
  __syncthreads();

  const int n = blockIdx.x * 256 + tid;
  const float* base = in + (size_t)b * 3 * HW;
  float x0 = base[n], x1 = base[HW + n], x2 = base[2 * HW + n];

  int best = 0; float bestd = 3.4e38f;
#pragma unroll
  for (int j = 0; j < NC; ++j) {
    float d0 = x0 - pal[j * 4 + 0];
    float d1 = x1 - pal[j * 4 + 1];
    float d2c = x2 - pal[j * 4 + 2];
    float d = d0 * d0 + d1 * d1 + d2c * d2c;
    if (d < bestd) { bestd = d; best = j; }   // first-min tie break = argmin
  }
  float q0 = pal[best * 4 + 0];
  float q1 = pal[best * 4 + 1];
  float q2 = pal[best * 4 + 2];

  float* qb = out_q + (size_t)b * 3 * HW;
  qb[n] = q0; qb[HW + n] = q1; qb[2 * HW + n] = q2;

  float* eb = out_enc + (size_t)b * NC * HW;
#pragma unroll
  for (int j = 0; j < NC; ++j)
    eb[(size_t)j * HW + n] = (j == best) ? 1.0f : 0.0f;

  float err = (q0 - x0) * (q0 - x0) + (q1 - x1) * (q1 - x1) +
              (q2 - x2) * (q2 - x2);
  red[tid] = err;
  __syncthreads();
  for (int s = 128; s > 0; s >>= 1) {
    if (tid < s) red[tid] += red[tid + s];
    __syncthreads();
  }
  if (tid == 0) blockSums[b * gridDim.x + blockIdx.x] = red[0];
}

// ---------------------------------------------------------------- phase 4 ---
__global__ __launch_bounds__(256) void vq_loss_kernel(
    const float* __restrict__ blockSums, int nblocks,
    float* __restrict__ out_loss) {
  __shared__ float red[256];
  const int tid = threadIdx.x;
  float s = 0.0f;
  for (int i = tid; i < nblocks; i += 256) s += blockSums[i];
  red[tid] = s;
  __syncthreads();
  for (int st = 128; st > 0; st >>= 1) {
    if (tid < st) red[tid] += red[tid + st];
    __syncthreads();
  }
  if (tid == 0) {
    float mse = red[0] / (float)((size_t)BATCH * 3 * HW);
    out_loss[0] = 1.25f * mse;   // q_latent + 0.25 * e_latent (equal forward)
  }
}

// --------------------------------------------------------------------------
extern "C" void kernel_launch(void* const* d_in, const int* in_sizes, int n_in,
                              void* d_out, int out_size, void* d_ws,
                              size_t ws_size, hipStream_t stream) {
  const float* in  = (const float*)d_in[0];   // [16,3,256,256]
  const float* emb = (const float*)d_in[1];   // [256,3]
  float* out = (float*)d_out;

  // Flat output layout in return order:
  const size_t o_quant = 0;
  const size_t o_enc   = (size_t)BATCH * 3 * HW;            //  3,145,728
  const size_t o_pal   = o_enc + (size_t)BATCH * NC * HW;   // 19,922,944
  const size_t o_loss  = o_pal + (size_t)BATCH * NC * 3;    // 19,923,712

  float* ws        = (float*)d_ws;
  float* ws_dnorm2 = ws;                    // 4096 floats
  float* ws_pal    = ws + 4096;             // 1024 floats ([B][NC][4])
  float* ws_bsum   = ws + 4096 + 1024;      // 4096 floats

  vq_zero_kernel<<<16, 256, 0, stream>>>(ws_dnorm2, BATCH * KEMB);
  vq_dnorm2_kernel<<<dim3(P1_CHUNKS, BATCH), 256, 0, stream>>>(in, emb,
                                                               ws_dnorm2);
  vq_topk_kernel<<<BATCH, 256, 0, stream>>>(emb, ws_dnorm2, ws_pal,
                                            out + o_pal);
  vq_assign_kernel<<<dim3(HW / 256, BATCH), 256, 0, stream>>>(
      in, ws_pal, out + o_quant, out + o_enc, ws_bsum);
  vq_loss_kernel<<<1, 256, 0, stream>>>(ws_bsum, BATCH * (HW / 256),
                                        out + o_loss);
}